// Model_83519934038722
// MI455X (gfx1250) — compile-verified
//
#include <hip/hip_runtime.h>
#include <hip/hip_bf16.h>
#include <math.h>

// ---------------------------------------------------------------------------
// Problem constants (from the reference)
// ---------------------------------------------------------------------------
#define N_NODES 100000
#define N_EDGES 1600000
#define IN_C    128
#define HID_C   128
#define OUT_C   64
#define N_LAYERS 4
#define K_STEPS 10
#define ALPHA   0.1f
#define LN_EPS  1e-5f

typedef __attribute__((ext_vector_type(2))) float v2f;
typedef __attribute__((ext_vector_type(8))) float v8f;

// ---------------------------------------------------------------------------
// Helpers
// ---------------------------------------------------------------------------
__device__ __forceinline__ float wave_allreduce_sum(float v) {
    // wave32 butterfly: every lane ends with the full sum
    #pragma unroll
    for (int off = 16; off > 0; off >>= 1)
        v += __shfl_xor(v, off, 32);
    return v;
}

__device__ __forceinline__ float gelu_exact(float x) {
    // 0.5 * x * (1 + erf(x / sqrt(2)))
    return 0.5f * x * (1.0f + erff(x * 0.70710678118654752f));
}

// ---------------------------------------------------------------------------
// WMMA fp32 GEMM:  D[nrows x NCOLS] = A[nrows x 128] * W[NCOLS x 128]^T
// One block computes a 16-row stripe across all NCOLS columns.
// One wave per 16x16 output tile; K iterated in steps of 4 via
// V_WMMA_F32_16X16X4_F32 (exact fp32, matches the fp32 reference).
// ---------------------------------------------------------------------------
template <int NCOLS>
__global__ __launch_bounds__(32 * (NCOLS / 16))
void gemm_wmma_f32(const float* __restrict__ A,
                   const float* __restrict__ W,
                   float* __restrict__ D) {
    constexpr int NW = NCOLS / 16;
    __shared__ float At[16][132];   // 16x128 A tile, padded

    const int tid  = threadIdx.x;
    const int wave = tid >> 5;
    const int lane = tid & 31;
    const int row0 = blockIdx.x * 16;

    // Cooperative, coalesced load of the 16x128 A tile into LDS
    for (int i = tid; i < 16 * 32; i += NW * 32) {
        const int r  = i >> 5;
        const int c4 = i & 31;
        float4 v = ((const float4*)(A + (size_t)(row0 + r) * 128))[c4];
        At[r][c4 * 4 + 0] = v.x;
        At[r][c4 * 4 + 1] = v.y;
        At[r][c4 * 4 + 2] = v.z;
        At[r][c4 * 4 + 3] = v.w;
    }
    __syncthreads();

    // ISA 32-bit A/B fragment layout:
    //   lanes 0-15 hold K = kb+0 / kb+1 ; lanes 16-31 hold K = kb+2 / kb+3
    const int m    = lane & 15;
    const int koff = (lane >> 4) << 1;
    const int n    = wave * 16 + m;
    const float* wrow = W + (size_t)n * 128;   // W is [NCOLS x 128] row-major

    v8f acc = {};
    #pragma unroll
    for (int kb = 0; kb < 128; kb += 4) {
        v2f a, b;
        a.x = At[m][kb + koff];
        a.y = At[m][kb + koff + 1];
        b.x = wrow[kb + koff];
        b.y = wrow[kb + koff + 1];
        acc = __builtin_amdgcn_wmma_f32_16x16x4_f32(
            /*neg_a=*/false, a, /*neg_b=*/false, b,
            /*c_mod=*/(short)0, acc, /*reuse_a=*/false, /*reuse_b=*/false);
    }

    // C/D layout: VGPR r -> row m0 = r (lanes 0-15) / r+8 (lanes 16-31)
    const int orow = row0 + ((lane >> 4) << 3);
    const int ocol = wave * 16 + m;
    #pragma unroll
    for (int r = 0; r < 8; ++r)
        D[(size_t)(orow + r) * NCOLS + ocol] = acc[r];
}

// ---------------------------------------------------------------------------
// Row-wise LayerNorm / GELU (one wave32 per 128-wide row, float4 per lane)
//   MODE 0: out = gelu(layer_norm(in))      (encoder)
//   MODE 1: out = layer_norm(gelu(in))      (per-layer epilogue)
// ---------------------------------------------------------------------------
template <int MODE>
__global__ __launch_bounds__(256)
void rowop_ln_gelu(const float* __restrict__ in,
                   float* __restrict__ out,
                   const float* __restrict__ gamma,
                   const float* __restrict__ beta) {
    const int wave = threadIdx.x >> 5;
    const int lane = threadIdx.x & 31;
    const int row  = blockIdx.x * (blockDim.x >> 5) + wave;
    if (row >= N_NODES) return;

    float4 v = ((const float4*)(in + (size_t)row * HID_C))[lane];
    if (MODE == 1) {
        v.x = gelu_exact(v.x); v.y = gelu_exact(v.y);
        v.z = gelu_exact(v.z); v.w = gelu_exact(v.w);
    }

    const float mu = wave_allreduce_sum(v.x + v.y + v.z + v.w) * (1.0f / HID_C);
    float4 c = make_float4(v.x - mu, v.y - mu, v.z - mu, v.w - mu);
    const float var = wave_allreduce_sum(c.x * c.x + c.y * c.y +
                                         c.z * c.z + c.w * c.w) * (1.0f / HID_C);
    const float rs = rsqrtf(var + LN_EPS);

    const float4 g = ((const float4*)gamma)[lane];
    const float4 b = ((const float4*)beta)[lane];
    float4 y;
    y.x = g.x * c.x * rs + b.x;
    y.y = g.y * c.y * rs + b.y;
    y.z = g.z * c.z * rs + b.z;
    y.w = g.w * c.w * rs + b.w;
    if (MODE == 0) {
        y.x = gelu_exact(y.x); y.y = gelu_exact(y.y);
        y.z = gelu_exact(y.z); y.w = gelu_exact(y.w);
    }
    ((float4*)(out + (size_t)row * HID_C))[lane] = y;
}

// ---------------------------------------------------------------------------
// agg = ALPHA * x0   (elementwise init; avoids a separate zero + blend pass)
// ---------------------------------------------------------------------------
__global__ __launch_bounds__(256)
void scale_init(const float* __restrict__ x0, float* __restrict__ agg) {
    const size_t i = (size_t)blockIdx.x * blockDim.x + threadIdx.x;
    float4 v = ((const float4*)x0)[i];
    v.x *= ALPHA; v.y *= ALPHA; v.z *= ALPHA; v.w *= ALPHA;
    ((float4*)agg)[i] = v;
}

// ---------------------------------------------------------------------------
// Edge scatter: agg[dst] += (1-ALPHA) * w * z[src]
// One wave32 per edge; each lane carries 4 channels (float4 gather,
// 4x global_atomic_add_f32 scatter). Pure HBM/atomic-bandwidth bound.
// ---------------------------------------------------------------------------
__global__ __launch_bounds__(256)
void edge_scatter(const float* __restrict__ z,
                  const int* __restrict__ src,
                  const int* __restrict__ dst,
                  const float* __restrict__ w,
                  float* __restrict__ agg) {
    const int wave = threadIdx.x >> 5;
    const int lane = threadIdx.x & 31;
    const int e = blockIdx.x * (blockDim.x >> 5) + wave;
    if (e >= N_EDGES) return;

    const int   s  = src[e];
    const int   d  = dst[e];
    const float wt = w[e] * (1.0f - ALPHA);

    float4 v = ((const float4*)(z + (size_t)s * HID_C))[lane];
    float* ap = agg + (size_t)d * HID_C + lane * 4;
    atomicAdd(ap + 0, v.x * wt);
    atomicAdd(ap + 1, v.y * wt);
    atomicAdd(ap + 2, v.z * wt);
    atomicAdd(ap + 3, v.w * wt);
}

// ---------------------------------------------------------------------------
// Orchestration
// ---------------------------------------------------------------------------
extern "C" void kernel_launch(void* const* d_in, const int* in_sizes, int n_in,
                              void* d_out, int out_size, void* d_ws, size_t ws_size,
                              hipStream_t stream) {
    const float* x     = (const float*)d_in[0];
    const int*   eidx  = (const int*)d_in[1];     // [2, N_EDGES]
    const float* ew    = (const float*)d_in[2];
    const float* Wenc  = (const float*)d_in[3];   // [128, 128]
    const float* Wdec  = (const float*)d_in[4];   // [64, 128]
    const float* gamma = (const float*)d_in[5];
    const float* beta  = (const float*)d_in[6];
    float*       out   = (float*)d_out;           // [N_NODES, 64]

    const int* src = eidx;
    const int* dst = eidx + N_EDGES;

    float* ws     = (float*)d_ws;
    float* bufX0  = ws;                               // x0 / h (layer input)
    float* bufZ   = ws + (size_t)N_NODES * HID_C;     // ping
    float* bufAgg = ws + 2 * (size_t)N_NODES * HID_C; // pong

    const dim3 blk256(256), blk128(128);
    const int gemm_blocks = N_NODES / 16;             // 6250
    const int row_blocks  = N_NODES / 8;              // 12500 (8 rows/block)
    const int init_blocks = (N_NODES * HID_C / 4) / 256; // 12500
    const int edge_blocks = N_EDGES / 8;              // 200000 (8 edges/block)

    // ---- Encoder: h = gelu(LN(x @ Wenc^T)) -> bufX0
    gemm_wmma_f32<HID_C><<<gemm_blocks, blk256, 0, stream>>>(x, Wenc, bufZ);
    rowop_ln_gelu<0><<<row_blocks, blk256, 0, stream>>>(bufZ, bufX0, gamma, beta);

    // ---- 4 layers of APPNP(K=10) + gelu + LN
    float* pp[2] = {bufAgg, bufZ};                    // agg target for step k = pp[k&1]
    for (int layer = 0; layer < N_LAYERS; ++layer) {
        const float* zptr = bufX0;                    // z starts as x0
        for (int k = 0; k < K_STEPS; ++k) {
            float* aggptr = pp[k & 1];
            scale_init<<<init_blocks, blk256, 0, stream>>>(bufX0, aggptr);
            edge_scatter<<<edge_blocks, blk256, 0, stream>>>(zptr, src, dst, ew, aggptr);
            zptr = aggptr;
        }
        // after k=9 (odd), final z lives in bufZ; write back into bufX0
        rowop_ln_gelu<1><<<row_blocks, blk256, 0, stream>>>(bufZ, bufX0, gamma, beta);
    }

    // ---- Decoder: out = h @ Wdec^T
    gemm_wmma_f32<OUT_C><<<gemm_blocks, blk128, 0, stream>>>(bufX0, Wdec, out);
}